// BSplineBasis_21526376087733
// MI455X (gfx1250) — compile-verified
//
#include <hip/hip_runtime.h>

// KAN B-spline basis: x[N,512] × grid[512,12] -> bases[N, 512*8]
// ~576MB of HBM traffic => 25us floor @ 23.3TB/s. The dense Cox-de Boor
// recursion is VALU-bound (~165 issue slots/element), so this version uses
// the local de Boor algorithm: find the knot span m (g[m]<=x<g[m+1]), compute
// only the 4 nonzero cubic bases (17 mul/fma), zero-fill + windowed b128
// store. All divisions live in per-feature reciprocal tables in LDS; the
// span-dependent knots/reciprocals are gathered with one LDS address register
// plus immediate ds_load_b32 offsets.

typedef float v4f  __attribute__((ext_vector_type(4)));
typedef float v4fu __attribute__((ext_vector_type(4), aligned(4)));

namespace {
constexpr int F_IN    = 512;  // input features
constexpr int GCOLS   = 12;   // grid columns (grid_size + 2*order + 1)
constexpr int NB1     = 11;   // G-1 order-0 bases (generic path)
constexpr int OUTW    = 8;    // outputs per (n, feature)
constexpr int FB      = 64;   // features per block tile
constexpr int NBROWS  = 32;   // rows per block tile
constexpr int THREADS = 256;  // 8 wave32s
constexpr int RPT     = 8;    // rows per thread (NBROWS / (THREADS/FB))
constexpr int LSTR    = 43;   // LDS dword stride per feature (odd -> conflict-free)
constexpr int OG = 0;         // grid knots         [12]
constexpr int O1 = 12;        // 1/max(g[j+1]-g[j]) [11]
constexpr int O2 = 23;        // 1/max(g[j+2]-g[j]) [10]
constexpr int O3 = 33;        // 1/max(g[j+3]-g[j]) [9]
}  // namespace

// ---------------------------------------------------------------------------
// Fast path: requires nrows % NBROWS == 0 (true for the reference N = 32768).
// ---------------------------------------------------------------------------
__global__ __launch_bounds__(THREADS)
void kan_bspline_fast(const float* __restrict__ X,
                      const float* __restrict__ Grd,
                      float* __restrict__ Out)
{
    __shared__ float lds[FB * LSTR];  // 11008 B

    const int t  = threadIdx.x;
    const int f0 = blockIdx.x * FB;
    const int n0 = blockIdx.y * NBROWS;

    const int fl    = t & (FB - 1);   // feature within tile
    const int rg    = t >> 6;         // row group 0..3
    const int nbase = n0 + rg * RPT;
    const float* xp = X + (size_t)nbase * F_IN + (f0 + fl);

    // Warm WGP$/L2 for this tile's x while we build spline tables
    // (single clause of global_prefetch_b8 with immediate offsets).
#pragma unroll
    for (int rr = 0; rr < RPT; ++rr)
        __builtin_prefetch(xp + (size_t)rr * F_IN, 0, 3);

    // Phase 1: stage grid rows [f0, f0+FB) into LDS (coalesced global reads).
    for (int idx = t; idx < FB * GCOLS; idx += THREADS) {
        const int f = idx / GCOLS;
        const int j = idx - f * GCOLS;
        lds[f * LSTR + OG + j] = Grd[(size_t)f0 * GCOLS + idx];
    }
    __syncthreads();

    // Phase 2: per-feature reciprocal tables — the ONLY divides in the kernel.
    for (int id = t; id < FB * 30; id += THREADS) {
        const int f = id / 30;
        const int r = id - f * 30;
        float* base = &lds[f * LSTR];
        constexpr float eps = 1e-8f;
        if (r < 11) {
            const float d = base[OG + r + 1] - base[OG + r];
            base[O1 + r] = 1.0f / fmaxf(d, eps);
        } else if (r < 21) {
            const int j = r - 11;
            const float d = base[OG + j + 2] - base[OG + j];
            base[O2 + j] = 1.0f / fmaxf(d, eps);
        } else {
            const int j = r - 21;
            const float d = base[OG + j + 3] - base[OG + j];
            base[O3 + j] = 1.0f / fmaxf(d, eps);
        }
    }
    __syncthreads();

    // Knots register-resident for the span search (reciprocals stay in LDS
    // and are gathered per element -> low VGPR pressure, high occupancy).
    float g[GCOLS];
    {
        const int fb = fl * LSTR;
#pragma unroll
        for (int j = 0; j < GCOLS; ++j) g[j] = lds[fb + OG + j];
    }

    // Preload all 8 row x values (back-to-back NT b32 loads, one wait).
    float xs[RPT];
#pragma unroll
    for (int rr = 0; rr < RPT; ++rr)
        xs[rr] = __builtin_nontemporal_load(xp + (size_t)rr * F_IN);

    float* op = Out + (size_t)nbase * (F_IN * OUTW) + (size_t)(f0 + fl) * OUTW;
    const int fbase = fl * LSTR;
    const v4f zero = {0.0f, 0.0f, 0.0f, 0.0f};

#pragma unroll
    for (int rr = 0; rr < RPT; ++rr) {
        const float x = xs[rr];

        // Knot span m: g[m] <= x < g[m+1]  (m = #{j in 1..11 : x >= g[j]}).
        int mi = 0;
#pragma unroll
        for (int j = 1; j < GCOLS; ++j) mi += (x >= g[j]) ? 1 : 0;
        mi = (mi < 3) ? 3 : ((mi > 7) ? 7 : mi);  // memory-safety clamp; inert for x in-range

        // Gather 6 knots + 6 reciprocals: one LDS address, immediate offsets.
        const int a0 = fbase + (mi - 2);
        const float t0 = x - lds[a0 + 0];   // x - g[m-2]
        const float t1 = x - lds[a0 + 1];   // x - g[m-1]
        const float t2 = x - lds[a0 + 2];   // x - g[m]
        const float t3 = x - lds[a0 + 3];   // x - g[m+1]
        const float t4 = x - lds[a0 + 4];   // x - g[m+2]
        const float t5 = x - lds[a0 + 5];   // x - g[m+3]
        const float w1  = lds[a0 + O1 + 2]; // inv1[m]
        const float u2a = lds[a0 + O2 + 1]; // inv2[m-1]
        const float u2b = lds[a0 + O2 + 2]; // inv2[m]
        const float u3a = lds[a0 + O3 + 0]; // inv3[m-2]
        const float u3b = lds[a0 + O3 + 1]; // inv3[m-1]
        const float u3c = lds[a0 + O3 + 2]; // inv3[m]

        // Local de Boor (Piegl-Tiller A2.2): N[0..3] = B_{m-3..m,3}(x).
        // Same denominators (eps-guarded) as the reference recursion.
        float N0, N1, N2, N3, sv, tp;
        // degree 1
        N1 = t2 * w1;                       // saved
        N0 = -t3 * w1;
        // degree 2
        tp = N0 * u2a;  N0 = -t3 * tp;          sv = t1 * tp;
        tp = N1 * u2b;  N1 = sv - t4 * tp;      N2 = t2 * tp;
        // degree 3
        tp = N0 * u3a;  N0 = -t3 * tp;          sv = t0 * tp;
        tp = N1 * u3b;  N1 = sv - t4 * tp;      sv = t1 * tp;
        tp = N2 * u3c;  N2 = sv - t5 * tp;      N3 = t2 * tp;

        // Zero-fill the 8-slot record, then overwrite the 4-value window at
        // slot m-3 (dword-aligned b128; same-wave stores complete in order).
        float* o = op + (size_t)rr * F_IN * OUTW;
        __builtin_nontemporal_store(zero, (v4f*)o);
        __builtin_nontemporal_store(zero, (v4f*)o + 1);
        v4fu w = {N0, N1, N2, N3};
        __builtin_nontemporal_store(w, (v4fu*)(o + (mi - 3)));
    }
}

// ---------------------------------------------------------------------------
// Generic fallback: one (n, feature) element per thread, any nrows.
// Never launched for the reference shape; kept separate from the hot I$.
// ---------------------------------------------------------------------------
__global__ __launch_bounds__(THREADS)
void kan_bspline_generic(const float* __restrict__ X,
                         const float* __restrict__ Grd,
                         float* __restrict__ Out,
                         int nrows)
{
    const long long e = (long long)blockIdx.x * THREADS + threadIdx.x;
    const long long total = (long long)nrows * F_IN;
    if (e >= total) return;
    const int f = (int)(e % F_IN);
    const long long n = e / F_IN;

    const float x = X[e];
    const float* g = Grd + (size_t)f * GCOLS;

    constexpr float eps = 1e-8f;
    float b[NB1];
    for (int j = 0; j < NB1; ++j)
        b[j] = (x >= g[j] && x < g[j + 1]) ? 1.0f : 0.0f;
    for (int k = 1; k <= 3; ++k)
        for (int j = 0; j < NB1 - k; ++j) {
            const float l = (x - g[j]) / fmaxf(g[j + k] - g[j], eps);
            const float r = (g[j + k + 1] - x) / fmaxf(g[j + k + 1] - g[j + 1], eps);
            b[j] = l * b[j] + r * b[j + 1];
        }

    float* o = Out + (size_t)n * (F_IN * OUTW) + (size_t)f * OUTW;
    for (int j = 0; j < OUTW; ++j) o[j] = b[j];
}

extern "C" void kernel_launch(void* const* d_in, const int* in_sizes, int n_in,
                              void* d_out, int out_size, void* d_ws, size_t ws_size,
                              hipStream_t stream) {
    const float* x    = (const float*)d_in[0];   // [N, 512] f32
    const float* grid = (const float*)d_in[1];   // [512, 12] f32
    float* out        = (float*)d_out;           // [N, 4096] f32

    const int nrows = in_sizes[0] / F_IN;
    if (nrows % NBROWS == 0) {
        dim3 blk(THREADS);
        dim3 grd(F_IN / FB, nrows / NBROWS);
        kan_bspline_fast<<<grd, blk, 0, stream>>>(x, grid, out);
    } else {
        const long long total = (long long)nrows * F_IN;
        const int nblk = (int)((total + THREADS - 1) / THREADS);
        kan_bspline_generic<<<nblk, THREADS, 0, stream>>>(x, grid, out, nrows);
    }
}